// GraphConvolutionDiagLayer_73469710566062
// MI455X (gfx1250) — compile-verified
//
#include <hip/hip_runtime.h>
#include <cstdint>

// ---------------------------------------------------------------------------
// GraphConvolutionDiag: out[d] += ew[e] * (x[src[e]] * W)   (f32, atomic scatter)
// Strategy: wave-per-edge (32 lanes x float4 = 128-float row), edge metadata
// streamed Global->LDS with CDNA5 async loads (ASYNCcnt), double buffered.
// ---------------------------------------------------------------------------

#define F_DIM 128
#define TILE 1024
#define BLOCK 256
#define WAVES_PER_BLOCK (BLOCK / 32)
#define EDGES_PER_WAVE (TILE / WAVES_PER_BLOCK) // 128

#ifdef __has_builtin
#if __has_builtin(__builtin_amdgcn_global_load_async_to_lds_b128)
#define USE_ASYNC_LDS 1
#endif
#endif

#define AS1 __attribute__((address_space(1)))
#define AS3 __attribute__((address_space(3)))

typedef int v4i __attribute__((vector_size(4 * sizeof(int))));
typedef AS1 v4i* gptr_v4i;   // global: 64-bit pointer
typedef AS3 v4i* lptr_v4i;   // LDS: 32-bit offset pointer

__device__ __forceinline__ void async_copy_b128(const void* gsrc, void* ldst) {
#ifdef USE_ASYNC_LDS
  // per-lane 16B copy from global address gsrc to LDS address ldst; tracked by ASYNCcnt
  __builtin_amdgcn_global_load_async_to_lds_b128(
      (gptr_v4i)(uintptr_t)gsrc,
      (lptr_v4i)(uint32_t)(uintptr_t)ldst,
      /*offset=*/0, /*cpol=*/0);
#else
  *(float4*)ldst = *(const float4*)gsrc; // synchronous fallback
#endif
}

__device__ __forceinline__ void wait_async0() {
#ifdef USE_ASYNC_LDS
#if __has_builtin(__builtin_amdgcn_s_wait_asynccnt)
  __builtin_amdgcn_s_wait_asynccnt(0);
#else
  asm volatile("s_wait_asynccnt 0" ::: "memory");
#endif
#endif
}

__device__ __forceinline__ void process_edge(const float* __restrict__ x,
                                             float* __restrict__ out,
                                             float4 w4, int lane,
                                             int s, int d, float c) {
  const float4 xv = ((const float4*)(x + (size_t)s * F_DIM))[lane];
  float4 v;
  v.x = xv.x * w4.x * c;
  v.y = xv.y * w4.y * c;
  v.z = xv.z * w4.z * c;
  v.w = xv.w * w4.w * c;
  float* op = out + (size_t)d * F_DIM + (lane << 2);
  // relaxed device-scope fetch_add with unused result -> no-return
  // global_atomic_add_f32 (STOREcnt-tracked fast path through L2 atomics)
  __hip_atomic_fetch_add(op + 0, v.x, __ATOMIC_RELAXED, __HIP_MEMORY_SCOPE_AGENT);
  __hip_atomic_fetch_add(op + 1, v.y, __ATOMIC_RELAXED, __HIP_MEMORY_SCOPE_AGENT);
  __hip_atomic_fetch_add(op + 2, v.z, __ATOMIC_RELAXED, __HIP_MEMORY_SCOPE_AGENT);
  __hip_atomic_fetch_add(op + 3, v.w, __ATOMIC_RELAXED, __HIP_MEMORY_SCOPE_AGENT);
}

__global__ void __launch_bounds__(BLOCK)
gconv_edges_kernel(const float* __restrict__ x, const float* __restrict__ W,
                   const float* __restrict__ ew, const int* __restrict__ esrc,
                   const int* __restrict__ edst, float* __restrict__ out,
                   int n_edges, int num_tiles) {
  __shared__ int   s_src[2][TILE];
  __shared__ int   s_dst[2][TILE];
  __shared__ float s_wgt[2][TILE];

  const int tid  = threadIdx.x;
  const int lane = tid & 31;
  const int wave = tid >> 5;

  // per-lane slice of the diagonal weight (4 floats), loaded once
  const float4 w4 = ((const float4*)W)[lane];

  // ---------------- tiled body: async double-buffered edge streaming -------
  int t   = blockIdx.x;
  int buf = 0;
  if (t < num_tiles) {
    const int base = t * TILE + tid * 4;
    async_copy_b128(esrc + base, &s_src[0][tid * 4]);
    async_copy_b128(edst + base, &s_dst[0][tid * 4]);
    async_copy_b128(ew   + base, &s_wgt[0][tid * 4]);
  }
  for (; t < num_tiles; t += gridDim.x) {
    wait_async0();        // this wave's issued async loads have landed in LDS
    __syncthreads();      // all waves' slices of tile[buf] are visible

    const int tn = t + gridDim.x;
    if (tn < num_tiles) { // prefetch next tile into the other buffer
      const int base = tn * TILE + tid * 4;
      async_copy_b128(esrc + base, &s_src[buf ^ 1][tid * 4]);
      async_copy_b128(edst + base, &s_dst[buf ^ 1][tid * 4]);
      async_copy_b128(ew   + base, &s_wgt[buf ^ 1][tid * 4]);
    }

    const int jbeg = wave * EDGES_PER_WAVE;
#pragma unroll 4
    for (int j = 0; j < EDGES_PER_WAVE; ++j) {
      const int   s = s_src[buf][jbeg + j];  // same-address LDS broadcast
      const int   d = s_dst[buf][jbeg + j];
      const float c = s_wgt[buf][jbeg + j];
      process_edge(x, out, w4, lane, s, d, c);
    }

    __syncthreads();      // tile[buf] fully consumed before it is reused
    buf ^= 1;
  }

  // ---------------- remainder edges (E % TILE), direct path ----------------
  if (blockIdx.x == 0) {
    const int e0 = num_tiles * TILE;
    for (int e = e0 + wave; e < n_edges; e += WAVES_PER_BLOCK) {
      const int   s = esrc[e];
      const int   d = edst[e];
      const float c = ew[e];
      process_edge(x, out, w4, lane, s, d, c);
    }
  }
}

extern "C" void kernel_launch(void* const* d_in, const int* in_sizes, int n_in,
                              void* d_out, int out_size, void* d_ws,
                              size_t ws_size, hipStream_t stream) {
  (void)n_in; (void)d_ws; (void)ws_size;
  const float* x    = (const float*)d_in[0];
  const float* W    = (const float*)d_in[1];
  const float* ew   = (const float*)d_in[2];
  const int*   esrc = (const int*)d_in[3];
  const int*   edst = (const int*)d_in[4];
  float*       out  = (float*)d_out;

  const int n_edges   = in_sizes[2];
  const int num_tiles = n_edges / TILE;

  // output is accumulated with atomics -> must start from zero every call
  (void)hipMemsetAsync(d_out, 0, (size_t)out_size * sizeof(float), stream);

  int grid = num_tiles < 1 ? 1 : (num_tiles > 2048 ? 2048 : num_tiles);
  gconv_edges_kernel<<<grid, BLOCK, 0, stream>>>(x, W, ew, esrc, edst, out,
                                                 n_edges, num_tiles);
}